// AdaptiveGatingHybridActivation_43482248905184
// MI455X (gfx1250) — compile-verified
//
#include <hip/hip_runtime.h>
#include <hip/hip_bf16.h>

#define ALPHA_C 0.5f
#define BETA_C  0.1f
#define EPS_C   1e-10f
#define PIPE_D  8      // async software-pipeline depth (per-wave outstanding B128 chunks)
#define AHEAD_ROWS 192 // L2 prefetch distance in rows (~39MB in flight, L2 = 192MB)

// ---------- wave32 reductions ----------
__device__ __forceinline__ float warpRedSum(float v) {
#pragma unroll
  for (int o = 16; o > 0; o >>= 1) v += __shfl_xor(v, o, 32);
  return v;
}
__device__ __forceinline__ float warpRedMax(float v) {
#pragma unroll
  for (int o = 16; o > 0; o >>= 1) v = fmaxf(v, __shfl_xor(v, o, 32));
  return v;
}

// ---------- CDNA5 async global->LDS DMA (ASYNCcnt-tracked) ----------
// GVS mode: saddr = 64-bit SGPR base, vaddr = 32-bit byte offset,
// first operand = wave-relative LDS byte address. LU: line is dead after this read.
__device__ __forceinline__ void asyncLoadB32(unsigned ldsAddr, unsigned gOff,
                                             unsigned long long base) {
  asm volatile("global_load_async_to_lds_b32 %0, %1, %2 th:TH_LOAD_LU"
               :: "v"(ldsAddr), "v"(gOff), "s"(base) : "memory");
}
__device__ __forceinline__ void asyncLoadB128(unsigned ldsAddr, unsigned gOff,
                                              unsigned long long base) {
  asm volatile("global_load_async_to_lds_b128 %0, %1, %2 th:TH_LOAD_LU"
               :: "v"(ldsAddr), "v"(gOff), "s"(base) : "memory");
}
__device__ __forceinline__ void waitAsyncD() {   // allow PIPE_D outstanding
  asm volatile("s_wait_asynccnt 8" ::: "memory");
}
__device__ __forceinline__ void waitAsync0() {   // drain
  asm volatile("s_wait_asynccnt 0" ::: "memory");
}

// One workgroup per row. Row (201KB) staged in LDS via pipelined async DMA
// overlapped with pass A; pass B runs from LDS.
__global__ __launch_bounds__(1024, 1)
void row_loss_kernel(const float* __restrict__ logits,
                     const long long* __restrict__ targets,
                     float* __restrict__ rowLoss, int V) {
  extern __shared__ __align__(16) unsigned char s_raw[];
  __shared__ float sred[4][32];
  __shared__ float sbc[4];

  const int row  = blockIdx.x;
  const int tid  = threadIdx.x;
  const int nthr = blockDim.x;
  const int lane = tid & 31;
  const int wid  = tid >> 5;
  const int nw   = nthr >> 5;

  const float* rowp = logits + (size_t)row * (size_t)V;
  const unsigned long long gbase = (unsigned long long)rowp;
  const unsigned pad = (unsigned)(gbase & 15ull);            // 0,4,8,12
  const unsigned ldsBase = (unsigned)(unsigned long long)(size_t)s_raw;
  float* sx = (float*)(s_raw + pad);                         // LDS ≡ global (mod 16)

  int n0 = ((16 - (int)pad) & 15) >> 2;                      // prologue elems to 16B align
  if (n0 > V) n0 = V;
  const int nc = (V - n0) >> 2;                              // aligned 16B chunks
  const int tailStart = n0 + (nc << 2);

  // ---- issue misaligned prologue/tail first (in-order completion keeps counting simple)
  for (int i = tid; i < n0; i += nthr)
    asyncLoadB32(ldsBase + pad + 4u * (unsigned)i, 4u * (unsigned)i, gbase);
  for (int i = tailStart + tid; i < V; i += nthr)
    asyncLoadB32(ldsBase + pad + 4u * (unsigned)i, 4u * (unsigned)i, gbase);

  // ---- prime the pipeline: first PIPE_D chunks per thread
  int c_issue = tid;
#pragma unroll
  for (int d = 0; d < PIPE_D; ++d) {
    if (c_issue < nc) {
      const unsigned off = (unsigned)(n0 + 4 * c_issue) * 4u;
      asyncLoadB128(ldsBase + pad + off, off, gbase);
      c_issue += nthr;
    }
  }

  // ---- warm L2 for a future row while this row streams in
  if (row + AHEAD_ROWS < (int)gridDim.x) {
    const char* np = (const char*)(logits + (size_t)(row + AHEAD_ROWS) * (size_t)V);
    const int bytes = V * 4;
    for (int off = tid * 256; off < bytes; off += nthr * 256)
      __builtin_prefetch(np + off, 0, 0);                    // global_prefetch_b8
  }

  // ---- pass A fused with DMA: each thread consumes exactly what it loaded,
  //      in issue order; per-wave in-order ASYNCcnt makes s_wait_asynccnt D exact.
  float lsum = 0.f, lsq = 0.f, lmax = -3.402823466e38f;
  const float4* sx4 = (const float4*)(s_raw + pad + 4u * (unsigned)n0); // 16B-aligned
  for (int c = tid; c < nc; c += nthr) {
    if (c_issue < nc) {
      const unsigned off = (unsigned)(n0 + 4 * c_issue) * 4u;
      asyncLoadB128(ldsBase + pad + off, off, gbase);
      c_issue += nthr;
      waitAsyncD();                       // chunk c guaranteed complete
    } else {
      waitAsync0();                       // drain at the tail of the pipeline
    }
    const float4 v = sx4[c];
    lsum += v.x + v.y + v.z + v.w;
    lsq  = fmaf(v.x, v.x, lsq); lsq = fmaf(v.y, v.y, lsq);
    lsq  = fmaf(v.z, v.z, lsq); lsq = fmaf(v.w, v.w, lsq);
    lmax = fmaxf(lmax, fmaxf(fmaxf(v.x, v.y), fmaxf(v.z, v.w)));
  }
  waitAsync0();                           // prologue/tail b32s + any remainder
  if (tid < n0) {                         // same thread that loaded it
    const float x = sx[tid];
    lsum += x; lsq = fmaf(x, x, lsq); lmax = fmaxf(lmax, x);
  }
  for (int i = tailStart + tid; i < V; i += nthr) {
    const float x = sx[i];
    lsum += x; lsq = fmaf(x, x, lsq); lmax = fmaxf(lmax, x);
  }

  lsum = warpRedSum(lsum); lsq = warpRedSum(lsq); lmax = warpRedMax(lmax);
  if (lane == 0) { sred[0][wid] = lsum; sred[1][wid] = lsq; sred[2][wid] = lmax; }
  __syncthreads();                        // also: every wave drained -> full row in LDS
  if (wid == 0) {
    float a = (lane < nw) ? sred[0][lane] : 0.f;
    float b = (lane < nw) ? sred[1][lane] : 0.f;
    float m = (lane < nw) ? sred[2][lane] : -3.402823466e38f;
    a = warpRedSum(a); b = warpRedSum(b); m = warpRedMax(m);
    if (lane == 0) {
      const float mean = a / (float)V;
      float var = (b - a * mean) / (float)(V - 1);           // unbiased (ddof=1)
      var = fmaxf(var, 0.f);
      const float stdv = sqrtf(var);
      sbc[0] = mean;
      sbc[1] = 1.f / (stdv + EPS_C);                         // 1/(std+eps)
      sbc[2] = 1.f / (1.f + BETA_C * stdv);                  // 1/T
      sbc[3] = m;                                            // row max
    }
  }
  __syncthreads();
  const float mean = sbc[0], invStd = sbc[1], invT = sbc[2], xmax = sbc[3];

  // ---- pass B from LDS: Z=sum e, S1=sum g*rp, S2=sum g*e, SR=sum g*log(g+eps)
  float z = 0.f, s1 = 0.f, s2 = 0.f, sreg = 0.f;
  auto accumB = [&](float x) {
    const float xm = x - mean;
    const float e  = __expf((x - xmax) * invT);
    const float g  = 1.f / (1.f + __expf(-xm * invStd));     // sigmoid gate
    const float sc = __logf(1.f + fmaxf(xm, 0.f));           // log1p(relu(xm))
    const float e2 = __expf(2.f * (x / (1.f + sc)));
    const float rp = 1.f - 1.f / (e2 + 1.f);                 // (tanh+1)/2, overflow-safe
    z += e;
    s1 = fmaf(g, rp, s1);
    s2 = fmaf(g, e, s2);
    sreg = fmaf(g, __logf(g + EPS_C), sreg);
  };
  for (int c = tid; c < nc; c += nthr) {
    const float4 v = sx4[c];
    accumB(v.x); accumB(v.y); accumB(v.z); accumB(v.w);
  }
  if (tid < n0) accumB(sx[tid]);
  for (int i = tailStart + tid; i < V; i += nthr) accumB(sx[i]);

  z = warpRedSum(z); s1 = warpRedSum(s1); s2 = warpRedSum(s2); sreg = warpRedSum(sreg);
  if (lane == 0) { sred[0][wid] = z; sred[1][wid] = s1; sred[2][wid] = s2; sred[3][wid] = sreg; }
  __syncthreads();

  if (tid == 0) {
    float Z = 0.f, S1 = 0.f, S2 = 0.f, SR = 0.f;
    for (int w = 0; w < nw; ++w) { Z += sred[0][w]; S1 += sred[1][w]; S2 += sred[2][w]; SR += sred[3][w]; }
    // sum(combined) = a*S1 + sum(soft) - a*S2/Z, with sum(soft) = 1
    const float sumc = ALPHA_C * S1 + 1.f - ALPHA_C * (S2 / Z);

    const long long tgt = targets[row];
    const float xt  = sx[(int)tgt];
    const float xm  = xt - mean;
    const float et  = __expf((xt - xmax) * invT);
    const float softT = et / Z;
    const float gT  = 1.f / (1.f + __expf(-xm * invStd));
    const float scT = __logf(1.f + fmaxf(xm, 0.f));
    const float e2T = __expf(2.f * (xt / (1.f + scT)));
    const float rpT = 1.f - 1.f / (e2T + 1.f);
    const float combT = ALPHA_C * gT * rpT + (1.f - ALPHA_C * gT) * softT;
    float prob = combT / (sumc + EPS_C);
    prob = fminf(fmaxf(prob, EPS_C), 1.f);
    rowLoss[row] = -__logf(prob) + 0.01f * SR;
  }
}

// Deterministic final mean over row losses (no atomics).
__global__ __launch_bounds__(256, 1)
void mean_reduce_kernel(const float* __restrict__ rowLoss, int rows,
                        float* __restrict__ out) {
  __shared__ float s[8];
  float v = 0.f;
  for (int i = threadIdx.x; i < rows; i += blockDim.x) v += rowLoss[i];
  v = warpRedSum(v);
  const int lane = threadIdx.x & 31, wid = threadIdx.x >> 5;
  if (lane == 0) s[wid] = v;
  __syncthreads();
  if (threadIdx.x == 0) {
    float t = 0.f;
    const int nw = blockDim.x >> 5;
    for (int w = 0; w < nw; ++w) t += s[w];
    out[0] = t / (float)rows;
  }
}

extern "C" void kernel_launch(void* const* d_in, const int* in_sizes, int n_in,
                              void* d_out, int out_size, void* d_ws, size_t ws_size,
                              hipStream_t stream) {
  const float*     logits  = (const float*)d_in[0];
  const long long* targets = (const long long*)d_in[1];  // int64 targets
  const int rows = in_sizes[1];                          // B*S = 2048
  const int V    = in_sizes[0] / rows;                   // 50257
  float* rowLoss = (float*)d_ws;

  const size_t ldsBytes = (size_t)V * sizeof(float) + 16; // ~201KB of the 320KB WGP LDS
  (void)hipFuncSetAttribute(reinterpret_cast<const void*>(row_loss_kernel),
                            hipFuncAttributeMaxDynamicSharedMemorySize,
                            (int)ldsBytes);

  row_loss_kernel<<<rows, 1024, ldsBytes, stream>>>(logits, targets, rowLoss, V);
  mean_reduce_kernel<<<1, 256, 0, stream>>>(rowLoss, rows, (float*)d_out);
}